// FactorLayer_17033840296170
// MI455X (gfx1250) — compile-verified
//
#include <hip/hip_runtime.h>
#include <math.h>

#define DD 4096
#define BB 4096
#define TT 16
#define FEPS 1e-6f

typedef __attribute__((ext_vector_type(2))) float v2f;
typedef __attribute__((ext_vector_type(8))) float v8f;
typedef __attribute__((ext_vector_type(4))) unsigned int v4u;
typedef __attribute__((ext_vector_type(8))) int v8i;
typedef __attribute__((ext_vector_type(4))) int v4i;

// ---------------------------------------------------------------------------
// Kernel 1: per-column partial sums / sums of squares of `inputs`
// ---------------------------------------------------------------------------
__global__ __launch_bounds__(256) void stats_kernel(const float* __restrict__ in,
                                                    float* __restrict__ ps,
                                                    float* __restrict__ pq) {
  const int col = blockIdx.x * 256 + threadIdx.x;
  const int r0  = blockIdx.y * 64;
  const float* p = in + (size_t)r0 * DD + col;
  float s = 0.f, q = 0.f;
#pragma unroll 4
  for (int r = 0; r < 64; ++r) {
    float v = p[(size_t)r * DD];
    s += v;
    q += v * v;
  }
  ps[blockIdx.y * DD + col] = s;
  pq[blockIdx.y * DD + col] = q;
}

// ---------------------------------------------------------------------------
// Kernel 2: finalize mean/rstd; build Wq = rstd[d]*[W_static|W] repacked into
// WMMA B-operand order so the GEMM can load B as one b64 per lane:
//   element (K = 4*kb + j + 2*hi, N = 16*t + lo)  lives at
//   Wq[((kb*2 + t)*2 + hi)*32 + 2*lo + j]
// ---------------------------------------------------------------------------
__global__ __launch_bounds__(256) void finalize_kernel(const float* __restrict__ ps,
                                                       const float* __restrict__ pq,
                                                       const float* __restrict__ Wst,
                                                       const float* __restrict__ W,
                                                       float* __restrict__ meanp,
                                                       float* __restrict__ rstdp,
                                                       float* __restrict__ Wq) {
  const int d = blockIdx.x * 256 + threadIdx.x;
  float S = 0.f, Q = 0.f;
  for (int rb = 0; rb < 64; ++rb) {
    S += ps[rb * DD + d];
    Q += pq[rb * DD + d];
  }
  const float mean = S / (float)BB;
  const float var  = Q / (float)BB - mean * mean;
  const float rstd = 1.f / (sqrtf(fmaxf(var, 0.f)) + FEPS);
  meanp[d] = mean;
  rstdp[d] = rstd;

  const int kb = d >> 2;
  const int hi = (d >> 1) & 1;
  const int j  = d & 1;
#pragma unroll
  for (int t = 0; t < 2; ++t) {
    const size_t base = (size_t)((kb * 2 + t) * 2 + hi) * 32 + j;
    const float* src = (t == 0) ? Wst : W;
#pragma unroll
    for (int lo = 0; lo < 16; ++lo) {
      Wq[base + 2 * lo] = rstd * src[d * 16 + lo];
    }
  }
}

// ---------------------------------------------------------------------------
// Kernel 3: bias[t] = sum_d mean[d]*rstd[d]*Wcat[d][t]   (32 values)
// ---------------------------------------------------------------------------
__global__ __launch_bounds__(256) void bias_kernel(const float* __restrict__ meanp,
                                                   const float* __restrict__ rstdp,
                                                   const float* __restrict__ Wst,
                                                   const float* __restrict__ W,
                                                   float* __restrict__ bias) {
  __shared__ float acc[8][32];
  const int t = threadIdx.x & 31;
  const int g = threadIdx.x >> 5;
  float p = 0.f;
  for (int c = g; c < DD; c += 8) {
    const float mr = meanp[c] * rstdp[c];
    const float w  = (t < 16) ? Wst[c * 16 + t] : W[c * 16 + (t - 16)];
    p += mr * w;
  }
  acc[g][t] = p;
  __syncthreads();
  if (g == 0) {
    float b = 0.f;
#pragma unroll
    for (int gg = 0; gg < 8; ++gg) b += acc[gg][t];
    bias[t] = b;
  }
}

// ---------------------------------------------------------------------------
// TDM: async-copy one 16-row x 64-col f32 tile of `inputs` into LDS.
// data_size=4B; pad 4 DWORDs after every 64 DWORDs -> LDS row stride = 68
// floats, making the A-operand ds_load_b64 bank-conflict-free
// (lane banks 4*lo + {0..3} cover all 64 banks).
// Group0: count=1 | lds_addr | global_addr(57b) | type=2.
// Group1: dims: tensor 4096x4096, tile 64x16, dim0_stride 4096.
// ---------------------------------------------------------------------------
__device__ __forceinline__ void tdm_load_tile(const float* gptr, unsigned lds_off) {
  const unsigned long long ga = (unsigned long long)(uintptr_t)gptr;
  v4u g0;
  g0[0] = 1u;                                        // count=1, no gather
  g0[1] = lds_off;                                   // LDS byte address
  g0[2] = (unsigned)(ga & 0xFFFFFFFFull);            // global_addr[31:0]
  g0[3] = (unsigned)((ga >> 32) & 0x01FFFFFFull)     // global_addr[56:32]
        | (2u << 30);                                // type = 2 (image)
  v8i g1;
  g1[0] = (int)0x07520000u;  // data_size=2(4B), pad_en, interval=64DW, amount=4DW
  g1[1] = (int)(0x1000u << 16);   // tensor_dim0 = 4096 (low 16 bits)
  g1[2] = (int)(0x1000u << 16);   // tensor_dim0 hi=0 ; tensor_dim1 = 4096 (low 16)
  g1[3] = (int)(64u << 16);       // tensor_dim1 hi=0 ; tile_dim0 = 64
  g1[4] = 16;                     // tile_dim1 = 16 ; tile_dim2 = 0
  g1[5] = 4096;                   // tensor_dim0_stride[31:0]
  g1[6] = 0;                      // stride0 hi ; tensor_dim1_stride lo (unused)
  g1[7] = 0;
  const v4i z4 = {0, 0, 0, 0};
  const v8i z8 = {0, 0, 0, 0, 0, 0, 0, 0};
  __builtin_amdgcn_tensor_load_to_lds(g0, g1, z4, z4, z8, 0);
}

// ---------------------------------------------------------------------------
// Kernel 4: Yall = inputs @ W' - bias  via V_WMMA_F32_16X16X4_F32.
// One wave per 16-row M-tile, both 16-wide N-tiles. A is staged by TDM into
// double-buffered LDS (overlap copy with compute); B comes as b64 loads from
// the repacked Wq (512 KB, L2-resident).
// A layout: lane L: row m0+(L&15), K = 2*(L>>4)+{0,1}.
// C layout: VGPR v: row m0+v+8*(L>>4), col (L&15).
// ---------------------------------------------------------------------------
__global__ __launch_bounds__(32) void gemm_kernel(const float* __restrict__ in,
                                                  const float* __restrict__ Wq,
                                                  const float* __restrict__ bias,
                                                  float* __restrict__ Yall) {
  __shared__ float lds[2][16 * 68];
  const int lane = threadIdx.x;
  const int lo = lane & 15;
  const int hi = lane >> 4;
  const int m0 = blockIdx.x * 16;
  const unsigned lds0 = (unsigned)(unsigned long long)(uintptr_t)&lds[0][0];
  const unsigned lds1 = (unsigned)(unsigned long long)(uintptr_t)&lds[1][0];
  const float* gbase = in + (size_t)m0 * DD;

  tdm_load_tile(gbase + 0, lds0);
  tdm_load_tile(gbase + 64, lds1);

  v8f c0 = {};
  v8f c1 = {};

  const int NT = DD / 64;  // 64 K-tiles
  for (int t = 0; t < NT; ++t) {
    if (t < NT - 1) __builtin_amdgcn_s_wait_tensorcnt(1);   // tile t landed
    else            __builtin_amdgcn_s_wait_tensorcnt(0);

    const float* abuf = &lds[t & 1][lo * 68 + 2 * hi];
    const float* bptr = Wq + ((size_t)(t * 16) * 4 + hi) * 32 + 2 * lo;

#pragma unroll
    for (int kk = 0; kk < 16; ++kk) {
      v2f a  = *reinterpret_cast<const v2f*>(abuf + kk * 4);
      v2f b0 = *reinterpret_cast<const v2f*>(bptr + kk * 128);
      v2f b1 = *reinterpret_cast<const v2f*>(bptr + kk * 128 + 64);
      c0 = __builtin_amdgcn_wmma_f32_16x16x4_f32(false, a, false, b0, (short)0, c0, false, false);
      c1 = __builtin_amdgcn_wmma_f32_16x16x4_f32(false, a, false, b1, (short)0, c1, false, false);
    }

    if (t + 2 < NT) {
      asm volatile("s_wait_dscnt 0" ::: "memory");  // buffer fully consumed
      tdm_load_tile(gbase + (t + 2) * 64, (t & 1) ? lds1 : lds0);
    }
  }

  const float bv0 = bias[lo];
  const float bv1 = bias[16 + lo];
#pragma unroll
  for (int v = 0; v < 8; ++v) {
    const int row = m0 + v + 8 * hi;
    Yall[(size_t)row * 32 + lo]      = c0[v] - bv0;
    Yall[(size_t)row * 32 + 16 + lo] = c1[v] - bv1;
  }
}

// ---------------------------------------------------------------------------
// Kernel 5: sequential factor recurrence in rank space (single block).
// ---------------------------------------------------------------------------
__device__ __forceinline__ float waveSum(float v) {
#pragma unroll
  for (int m = 16; m >= 1; m >>= 1) v += __shfl_xor(v, m, 32);
  return v;
}

__global__ __launch_bounds__(1024) void factor_kernel(const float* __restrict__ Yall,
                                                      float* __restrict__ out) {
  __shared__ float sd[32][33];
  __shared__ float dots[36];
  const int tid  = threadIdx.x;
  const int lane = tid & 31;
  const int wv   = tid >> 5;

  float nmat[TT][4];
  float c[4];

  // ---- step 0 ----
  {
    float s = 0.f, q = 0.f;
#pragma unroll
    for (int kk = 0; kk < 4; ++kk) {
      const int r = tid + 1024 * kk;
      c[kk] = Yall[(size_t)r * 32 + 0];
      const float o = Yall[(size_t)r * 32 + 16];
      out[(size_t)r * 16 + 0] = o;
      s += c[kk];
      q += c[kk] * c[kk];
    }
    s = waveSum(s);
    q = waveSum(q);
    if (lane == 0) { sd[0][wv] = s; sd[1][wv] = q; }
    __syncthreads();
    if (tid == 0) {
      float S = 0.f, Q = 0.f;
      for (int w = 0; w < 32; ++w) { S += sd[0][w]; Q += sd[1][w]; }
      const float mean = S / (float)BB;
      const float var  = Q / (float)BB - mean * mean;
      dots[32] = mean;
      dots[33] = 1.f / (sqrtf(fmaxf(var, 0.f)) + FEPS);
    }
    __syncthreads();
    const float mean = dots[32], rs = dots[33];
#pragma unroll
    for (int kk = 0; kk < 4; ++kk) nmat[0][kk] = (c[kk] - mean) * rs;
    __syncthreads();
  }

  // ---- steps 1..15 ----
#pragma unroll
  for (int i = 1; i < TT; ++i) {
    float ys[4], y[4];
#pragma unroll
    for (int kk = 0; kk < 4; ++kk) {
      const int r = tid + 1024 * kk;
      ys[kk] = Yall[(size_t)r * 32 + i];
      y[kk]  = Yall[(size_t)r * 32 + 16 + i];
    }
#pragma unroll
    for (int j = 0; j < i; ++j) {
      float p1 = 0.f, p2 = 0.f;
#pragma unroll
      for (int kk = 0; kk < 4; ++kk) {
        p1 += nmat[j][kk] * ys[kk];
        p2 += nmat[j][kk] * y[kk];
      }
      p1 = waveSum(p1);
      p2 = waveSum(p2);
      if (lane == 0) { sd[2 * j][wv] = p1; sd[2 * j + 1][wv] = p2; }
    }
    __syncthreads();
    if (tid < 2 * i) {
      float v = 0.f;
      for (int w = 0; w < 32; ++w) v += sd[tid][w];
      dots[tid] = v * (1.f / (float)BB);
    }
    __syncthreads();

    float s = 0.f, q = 0.f;
#pragma unroll
    for (int kk = 0; kk < 4; ++kk) {
      float cc = ys[kk], oo = y[kk];
#pragma unroll
      for (int j = 0; j < i; ++j) {
        cc -= nmat[j][kk] * dots[2 * j];
        oo -= nmat[j][kk] * dots[2 * j + 1];
      }
      c[kk] = cc;
      const int r = tid + 1024 * kk;
      out[(size_t)r * 16 + i] = oo;
      s += cc;
      q += cc * cc;
    }
    __syncthreads();

    if (i < TT - 1) {
      s = waveSum(s);
      q = waveSum(q);
      if (lane == 0) { sd[0][wv] = s; sd[1][wv] = q; }
      __syncthreads();
      if (tid == 0) {
        float S = 0.f, Q = 0.f;
        for (int w = 0; w < 32; ++w) { S += sd[0][w]; Q += sd[1][w]; }
        const float mean = S / (float)BB;
        const float var  = Q / (float)BB - mean * mean;
        dots[32] = mean;
        dots[33] = 1.f / (sqrtf(fmaxf(var, 0.f)) + FEPS);
      }
      __syncthreads();
      const float mean = dots[32], rs = dots[33];
#pragma unroll
      for (int kk = 0; kk < 4; ++kk) nmat[i][kk] = (c[kk] - mean) * rs;
      __syncthreads();
    }
  }
}

// ---------------------------------------------------------------------------
extern "C" void kernel_launch(void* const* d_in, const int* in_sizes, int n_in,
                              void* d_out, int out_size, void* d_ws, size_t ws_size,
                              hipStream_t stream) {
  (void)in_sizes; (void)n_in; (void)out_size; (void)ws_size;
  const float* in  = (const float*)d_in[0];   // inputs   (4096 x 4096)
  const float* W   = (const float*)d_in[1];   // W        (4096 x 16)
  const float* Wst = (const float*)d_in[2];   // W_static (4096 x 16)
  float* out = (float*)d_out;

  float* ws    = (float*)d_ws;
  float* ps    = ws;                      // 64*4096
  float* pq    = ps + 64 * DD;            // 64*4096
  float* meanp = pq + 64 * DD;            // 4096
  float* rstdp = meanp + DD;              // 4096
  float* Wq    = rstdp + DD;              // 4096*32 (WMMA-operand packed)
  float* bias  = Wq + (size_t)DD * 32;    // 32
  float* Yall  = bias + 32;               // 4096*32   (total ~3.0 MB)

  stats_kernel<<<dim3(16, 64), 256, 0, stream>>>(in, ps, pq);
  finalize_kernel<<<16, 256, 0, stream>>>(ps, pq, Wst, W, meanp, rstdp, Wq);
  bias_kernel<<<1, 256, 0, stream>>>(meanp, rstdp, Wst, W, bias);
  gemm_kernel<<<DD / 16, 32, 0, stream>>>(in, Wq, bias, Yall);
  factor_kernel<<<1, 1024, 0, stream>>>(Yall, out);
}